// ResidualVQ_40896678592817
// MI455X (gfx1250) — compile-verified
//
#include <hip/hip_runtime.h>

typedef float v2f __attribute__((ext_vector_type(2)));
typedef float v8f __attribute__((ext_vector_type(8)));

#define NB  8      // batch
#define NT  4096   // time
#define NH  256    // hidden
#define ND  128    // codebook dim
#define NK  4      // num codebooks
#define NCS 1024   // codebook size
#define TSTRIDE 132  // LDS tile row stride in floats (128 + 4 pad -> conflict-free ds_load_b64)

#define WMMA_F32(c, a, b) \
  (c) = __builtin_amdgcn_wmma_f32_16x16x4_f32(false, (a), false, (b), (short)0, (c), false, false)

// ---------------------------------------------------------------------------
// 0) Precompute 0.5*||codebook row||^2 for all K*CS codes; zero loss acc.
// ---------------------------------------------------------------------------
__global__ void vq_prep(const float* __restrict__ cb, float* __restrict__ cbn,
                        float* __restrict__ lacc) {
    int c = blockIdx.x * blockDim.x + threadIdx.x;
    if (c == 0) *lacc = 0.0f;
    if (c < NK * NCS) {
        const float* p = cb + (size_t)c * ND;
        float s = 0.0f;
        for (int d = 0; d < ND; ++d) s += p[d] * p[d];
        cbn[c] = 0.5f * s;
    }
}

// ---------------------------------------------------------------------------
// 1) pre_proj: residual[b,d,t] = sum_h W_pre[d,h]*z[b,h,t] + b_pre[d]
//    One wave -> 16(d) x 64(t) tile. Also zeroes qsum for its tile.
// ---------------------------------------------------------------------------
__global__ __launch_bounds__(128) void pre_proj(
    const float* __restrict__ z, const float* __restrict__ Wp,
    const float* __restrict__ bp, float* __restrict__ residual,
    float* __restrict__ qsum) {
    int wave = (blockIdx.x * blockDim.x + threadIdx.x) >> 5;
    int lane = threadIdx.x & 31;
    const int CT = NT / 64;          // 64 column groups
    int ct = wave % CT;
    int tmp = wave / CT;
    int rd = tmp % (ND / 16);        // 8 row tiles
    int b  = tmp / (ND / 16);        // 8 batches
    int t0 = ct * 64, d0 = rd * 16;
    int h = lane >> 4, n = lane & 15;

    v8f acc0 = {}, acc1 = {}, acc2 = {}, acc3 = {};
    for (int kk = 0; kk < NH / 4; ++kk) {
        int k0 = kk * 4;
        // A: W_pre tile, M=d row, K=h.  A[M][K]: lane(M=n), vgpr j -> K=k0+2h+j
        v2f a = *(const v2f*)&Wp[(size_t)(d0 + n) * NH + k0 + 2 * h];
        v2f bf;
        size_t zrow0 = ((size_t)b * NH + k0 + 2 * h) * NT;
        // B: z tile, K=h row, N=t.  lane(N=n), vgpr j -> K=k0+2h+j
        int t = t0 + n;
        bf.x = z[zrow0 + t];          bf.y = z[zrow0 + NT + t];
        WMMA_F32(acc0, a, bf);
        bf.x = z[zrow0 + t + 16];     bf.y = z[zrow0 + NT + t + 16];
        WMMA_F32(acc1, a, bf);
        bf.x = z[zrow0 + t + 32];     bf.y = z[zrow0 + NT + t + 32];
        WMMA_F32(acc2, a, bf);
        bf.x = z[zrow0 + t + 48];     bf.y = z[zrow0 + NT + t + 48];
        WMMA_F32(acc3, a, bf);
    }
    v8f* accs[4] = {&acc0, &acc1, &acc2, &acc3};
#pragma unroll
    for (int j = 0; j < 4; ++j) {
#pragma unroll
        for (int r = 0; r < 8; ++r) {
            int d = d0 + r + 8 * h;               // C/D: row M = r + 8*half
            int t = t0 + j * 16 + n;              // col N = n
            size_t off = ((size_t)b * ND + d) * NT + t;
            residual[off] = (*accs[j])[r] + bp[d];
            qsum[off] = 0.0f;
        }
    }
}

// ---------------------------------------------------------------------------
// Async-stage one 16x128 codebook tile into LDS (padded rows).
// Each of 128 threads moves 4 x 16B chunks via GLOBAL_LOAD_ASYNC_TO_LDS_B128.
// Tracked with ASYNCcnt (4 instructions per wave per tile, in-order done).
// ---------------------------------------------------------------------------
__device__ __forceinline__ void issue_tile_async(const float* __restrict__ cb,
                                                 int c0, float* ldsbase, int tid) {
#pragma unroll
    for (int p = 0; p < 4; ++p) {
        int chunk = tid + p * 128;           // 0..511 16B chunks (32 per row)
        int row   = chunk >> 5;
        int colf  = (chunk & 31) << 2;       // float offset within row
        const float* g = cb + (size_t)(c0 + row) * ND + colf;
        unsigned l = (unsigned)(uintptr_t)(ldsbase + row * TSTRIDE + colf);
        asm volatile("global_load_async_to_lds_b128 %0, %1, off"
                     :: "v"(l), "v"(g) : "memory");
    }
}

// ---------------------------------------------------------------------------
// 2) One VQ stage: scores = residual^T * cb^T - 0.5||c||^2, argmax per t,
//    gather, residual -= q, qsum += q, commit-loss accumulate.
//    One wave owns one (b, 16-t) tile; A fragments register-resident;
//    codebook tiles double-buffered in LDS via async DMA, shared by 4 waves.
// ---------------------------------------------------------------------------
__global__ __launch_bounds__(128) void vq_stage(
    const float* __restrict__ cb, const float* __restrict__ cbn,
    float* __restrict__ residual, float* __restrict__ qsum,
    float* __restrict__ idx_out, float* __restrict__ lacc) {
    __shared__ float sTile[2][16 * TSTRIDE];
    __shared__ int   sIdx[4][16];
    int tid   = threadIdx.x;
    int wslot = tid >> 5;
    int lane  = tid & 31;
    int wave  = blockIdx.x * 4 + wslot;          // 0 .. B*T/16-1 = 2047
    int b  = wave >> 8;                          // T/16 = 256 tiles per batch
    int t0 = (wave & 255) << 4;
    int h = lane >> 4, n = lane & 15;

    // Preload A tile: A[M=t][K=d], lane(M=n), vgpr j -> K = 4*kk + 2h + j
    v2f afrag[32];
#pragma unroll
    for (int kk = 0; kk < 32; ++kk) {
        size_t base = ((size_t)b * ND + kk * 4 + 2 * h) * NT + t0 + n;
        afrag[kk].x = residual[base];
        afrag[kk].y = residual[base + NT];
    }

    float bestv[8];
    int   besti[8];
#pragma unroll
    for (int r = 0; r < 8; ++r) { bestv[r] = -3.4e38f; besti[r] = 0; }

    // Prologue: stage tiles 0 and 1.
    issue_tile_async(cb, 0,  &sTile[0][0], tid);
    issue_tile_async(cb, 16, &sTile[1][0], tid);

    for (int i = 0; i < NCS / 16; ++i) {
        // Tile i landed when <=4 of this wave's asyncs remain (in-order done).
        if (i < NCS / 16 - 1) {
            asm volatile("s_wait_asynccnt 0x4" ::: "memory");
        } else {
            asm volatile("s_wait_asynccnt 0x0" ::: "memory");
        }
        __syncthreads();                       // tile i visible block-wide

        // B[K=d][N=c] from LDS: lane(N=n), vgpr j -> K = 4*kk + 2h + j
        const float* bt = &sTile[i & 1][0] + n * TSTRIDE + 2 * h;
        v8f acca = {}, accb = {};
#pragma unroll
        for (int kk = 0; kk < 32; kk += 2) {
            v2f bf0 = *(const v2f*)(bt + kk * 4);
            v2f bf1 = *(const v2f*)(bt + kk * 4 + 4);
            WMMA_F32(acca, afrag[kk],     bf0);
            WMMA_F32(accb, afrag[kk + 1], bf1);
        }
        v8f acc = acca + accb;

        int   c0  = i * 16;
        float sub = cbn[c0 + n];
        int   col = c0 + n;
#pragma unroll
        for (int r = 0; r < 8; ++r) {
            float s = acc[r] - sub;
            if (s > bestv[r]) { bestv[r] = s; besti[r] = col; }
        }

        __syncthreads();                       // everyone done reading buf i&1
        if (i + 2 < NCS / 16)
            issue_tile_async(cb, (i + 2) * 16, &sTile[i & 1][0], tid);
    }

    // Reduce (val,idx) across the 16 lanes of each half; first-index tiebreak.
#pragma unroll
    for (int r = 0; r < 8; ++r) {
        float v = bestv[r]; int i = besti[r];
        for (int m = 8; m >= 1; m >>= 1) {
            float v2 = __shfl_xor(v, m, 16);
            int   i2 = __shfl_xor(i, m, 16);
            if (v2 > v || (v2 == v && i2 < i)) { v = v2; i = i2; }
        }
        bestv[r] = v; besti[r] = i;
    }
    if (lane == 0 || lane == 16) {
#pragma unroll
        for (int r = 0; r < 8; ++r) sIdx[wslot][8 * h + r] = besti[r];
    }
    __syncthreads();

    if (lane < 16)
        idx_out[(size_t)b * NT + t0 + lane] = (float)sIdx[wslot][lane];

    // Gather + update: 16 t * 128 d elements, 64 per lane.
    float acc2 = 0.0f;
    for (int e = lane; e < 16 * ND; e += 32) {
        int tl = e & 15;
        int d  = e >> 4;
        float q = cb[(size_t)sIdx[wslot][tl] * ND + d];
        size_t off = ((size_t)b * ND + d) * NT + t0 + tl;
        float r = residual[off] - q;
        residual[off] = r;
        qsum[off] += q;
        acc2 += r * r;                 // commit loss: mean(new_residual^2)
    }
    for (int m = 16; m >= 1; m >>= 1) acc2 += __shfl_xor(acc2, m, 32);
    if (lane == 0) atomicAdd(lacc, acc2);
}

// ---------------------------------------------------------------------------
// 3) post_proj: out[b,h,t] = sum_d W_post[h,d]*qsum[b,d,t] + b_post[h]
// ---------------------------------------------------------------------------
__global__ __launch_bounds__(128) void post_proj(
    const float* __restrict__ qsum, const float* __restrict__ Wp,
    const float* __restrict__ bp, float* __restrict__ out) {
    int wave = (blockIdx.x * blockDim.x + threadIdx.x) >> 5;
    int lane = threadIdx.x & 31;
    const int CT = NT / 64;
    int ct = wave % CT;
    int tmp = wave / CT;
    int rh = tmp % (NH / 16);        // 16 row tiles
    int b  = tmp / (NH / 16);
    int t0 = ct * 64, h0 = rh * 16;
    int hf = lane >> 4, n = lane & 15;

    v8f acc0 = {}, acc1 = {}, acc2 = {}, acc3 = {};
    for (int kk = 0; kk < ND / 4; ++kk) {
        int k0 = kk * 4;
        v2f a = *(const v2f*)&Wp[(size_t)(h0 + n) * ND + k0 + 2 * hf];
        size_t qrow0 = ((size_t)b * ND + k0 + 2 * hf) * NT;
        int t = t0 + n;
        v2f bf;
        bf.x = qsum[qrow0 + t];          bf.y = qsum[qrow0 + NT + t];
        WMMA_F32(acc0, a, bf);
        bf.x = qsum[qrow0 + t + 16];     bf.y = qsum[qrow0 + NT + t + 16];
        WMMA_F32(acc1, a, bf);
        bf.x = qsum[qrow0 + t + 32];     bf.y = qsum[qrow0 + NT + t + 32];
        WMMA_F32(acc2, a, bf);
        bf.x = qsum[qrow0 + t + 48];     bf.y = qsum[qrow0 + NT + t + 48];
        WMMA_F32(acc3, a, bf);
    }
    v8f* accs[4] = {&acc0, &acc1, &acc2, &acc3};
#pragma unroll
    for (int j = 0; j < 4; ++j) {
#pragma unroll
        for (int r = 0; r < 8; ++r) {
            int hh = h0 + r + 8 * hf;
            int t  = t0 + j * 16 + n;
            out[((size_t)b * NH + hh) * NT + t] = (*accs[j])[r] + bp[hh];
        }
    }
}

// ---------------------------------------------------------------------------
// 4) finalize: loss = total_sq_sum / (B*D*T)
// ---------------------------------------------------------------------------
__global__ void finalize(const float* __restrict__ lacc, float* __restrict__ out_loss) {
    if (threadIdx.x == 0 && blockIdx.x == 0)
        *out_loss = *lacc * (1.0f / (float)((size_t)NB * ND * NT));
}

extern "C" void kernel_launch(void* const* d_in, const int* in_sizes, int n_in,
                              void* d_out, int out_size, void* d_ws, size_t ws_size,
                              hipStream_t stream) {
    (void)in_sizes; (void)n_in; (void)out_size; (void)ws_size;
    const float* z      = (const float*)d_in[0];
    const float* W_pre  = (const float*)d_in[1];
    const float* b_pre  = (const float*)d_in[2];
    const float* W_post = (const float*)d_in[3];
    const float* b_post = (const float*)d_in[4];
    const float* cbs    = (const float*)d_in[5];

    float* out = (float*)d_out;
    // d_out layout: out (B*H*T) | indices (K*B*T as float) | loss (1)
    float* residual = out;                          // alias: dead before post_proj writes
    float* idx_base = out + (size_t)NB * NH * NT;
    float* loss_out = idx_base + (size_t)NK * NB * NT;

    float* qsum = (float*)d_ws;                     // B*D*T floats
    float* cbn  = qsum + (size_t)NB * ND * NT;      // K*CS floats
    float* lacc = cbn + (size_t)NK * NCS;           // 1 float

    vq_prep<<<(NK * NCS + 255) / 256, 256, 0, stream>>>(cbs, cbn, lacc);

    // B * (D/16) * (T/64) = 4096 waves, 4 waves/block
    pre_proj<<<4096 / 4, 128, 0, stream>>>(z, W_pre, b_pre, residual, qsum);

    // B * T/16 = 2048 waves per stage, 4 waves/block
    for (int k = 0; k < NK; ++k) {
        vq_stage<<<2048 / 4, 128, 0, stream>>>(
            cbs + (size_t)k * NCS * ND, cbn + (size_t)k * NCS,
            residual, qsum, idx_base + (size_t)k * NB * NT, lacc);
    }

    // B * (H/16) * (T/64) = 8192 waves
    post_proj<<<8192 / 4, 128, 0, stream>>>(qsum, W_post, b_post, out);

    finalize<<<1, 1, 0, stream>>>(lacc, loss_out);
}